// LLMTripleAlignLoss_53867479826538
// MI455X (gfx1250) — compile-verified
//
#include <hip/hip_runtime.h>
#include <hip/hip_bf16.h>
#include <math.h>

#define IGNORE_IDX (-100)

constexpr int kB   = 256;
constexpr int kS   = 8192;
constexpr int kN   = kB * kS;          // 2,097,152 positions
constexpr int kNCH = kN / 4;           // 524,288 chunks of 4 positions
constexpr int kNB  = 1024;             // partial-reduction blocks
constexpr int kTPB = 256;

typedef float v2f __attribute__((ext_vector_type(2)));
typedef float v8f __attribute__((ext_vector_type(8)));

// ---------------- per-position math ----------------

__device__ __forceinline__ float focal3(float l0, float l1, float l2,
                                        int lab, float* cnt) {
  bool valid = (lab != IGNORE_IDX);
  int safe = valid ? lab : 0;
  float m  = fmaxf(l0, fmaxf(l1, l2));
  float e0 = __expf(l0 - m), e1 = __expf(l1 - m), e2 = __expf(l2 - m);
  float lse = m + __logf(e0 + e1 + e2);
  float lt = (safe == 0) ? l0 : ((safe == 1) ? l1 : l2);
  float ce = lse - lt;                 // -log_softmax[label]
  float pt = __expf(-ce);
  float om = 1.0f - pt;
  float fl = om * om * ce;             // gamma = 2
  if (valid) { *cnt += 1.0f; return fl; }
  return 0.0f;
}

__device__ __forceinline__ float bce_logit(float z, float y) {
  float t = __expf(-fabsf(z));
  return fmaxf(z, 0.0f) - z * y + __logf(1.0f + t);
}

// ---------------- pass 1: streaming partials ----------------
// ws layout: ws[a * kNB + block], a in {0:fl_a, 1:n_a, 2:fl_o, 3:n_o, 4:bce}

__global__ void __launch_bounds__(kTPB)
partial_kernel(const float* __restrict__ alog,
               const float* __restrict__ olog,
               const float* __restrict__ blog,
               const int*   __restrict__ alab,
               const int*   __restrict__ olab,
               float* __restrict__ ws) {
  float fa = 0.f, na = 0.f, fo = 0.f, no = 0.f, bs = 0.f;
  const int tid = threadIdx.x;
  const int gthreads = gridDim.x * blockDim.x;

  for (int c = blockIdx.x * blockDim.x + tid; c < kNCH; c += gthreads) {
    const int i0 = c * 4;
    // labels: int4 (b128) + one scalar for the cross-chunk neighbor
    int4 al = ((const int4*)alab)[c];
    int4 ol = ((const int4*)olab)[c];
    int anext, onext;
    if (((i0 + 3) & (kS - 1)) == (kS - 1)) { anext = -1; onext = -1; }
    else { anext = alab[i0 + 4]; onext = olab[i0 + 4]; }

    // logits: pure b128 streams
    const float4* ap = (const float4*)alog + 3 * (size_t)c;
    float4 a0 = ap[0], a1 = ap[1], a2 = ap[2];
    const float4* op = (const float4*)olog + 3 * (size_t)c;
    float4 o0 = op[0], o1 = op[1], o2 = op[2];
    const float4* bp = (const float4*)blog + 2 * (size_t)c;
    float4 b0 = bp[0], b1 = bp[1];

    float aL[12] = {a0.x,a0.y,a0.z,a0.w, a1.x,a1.y,a1.z,a1.w, a2.x,a2.y,a2.z,a2.w};
    float oL[12] = {o0.x,o0.y,o0.z,o0.w, o1.x,o1.y,o1.z,o1.w, o2.x,o2.y,o2.z,o2.w};
    float bL[8]  = {b0.x,b0.y,b0.z,b0.w, b1.x,b1.y,b1.z,b1.w};
    int   av[5]  = {al.x, al.y, al.z, al.w, anext};
    int   ov[5]  = {ol.x, ol.y, ol.z, ol.w, onext};

#pragma unroll
    for (int j = 0; j < 4; ++j) {
      fa += focal3(aL[3*j], aL[3*j+1], aL[3*j+2], av[j], &na);
      fo += focal3(oL[3*j], oL[3*j+1], oL[3*j+2], ov[j], &no);
      bool a_s = (av[j] == 1),  o_s = (ov[j] == 1);
      bool a_e = (av[j] == 2) && (av[j+1] != 2);
      bool o_e = (ov[j] == 2) && (ov[j+1] != 2);
      float y0 = (a_s | o_s) ? 1.0f : 0.0f;
      float y1 = (a_e | o_e) ? 1.0f : 0.0f;
      bs += bce_logit(bL[2*j], y0) + bce_logit(bL[2*j+1], y1);
    }
  }

  // wave32 tree reduce (fixed dataflow -> deterministic)
#pragma unroll
  for (int off = 16; off > 0; off >>= 1) {
    fa += __shfl_down(fa, off, 32);
    na += __shfl_down(na, off, 32);
    fo += __shfl_down(fo, off, 32);
    no += __shfl_down(no, off, 32);
    bs += __shfl_down(bs, off, 32);
  }
  __shared__ float red[kTPB / 32][5];
  const int wid = tid >> 5, lane = tid & 31;
  if (lane == 0) {
    red[wid][0] = fa; red[wid][1] = na; red[wid][2] = fo;
    red[wid][3] = no; red[wid][4] = bs;
  }
  __syncthreads();
  if (tid < 5) {
    float s = 0.f;
#pragma unroll
    for (int w = 0; w < kTPB / 32; ++w) s += red[w][tid];
    ws[tid * kNB + blockIdx.x] = s;   // store, not atomic: deterministic
  }
}

// ---------------- pass 2: WMMA reduction + scalar combine ----------------
// Sums each 1024-entry partial array with V_WMMA_F32_16X16X4_F32 against an
// all-ones B matrix: D = A*1 + C accumulates 64 f32 values per issue.
// A layout (ISA 16x4 f32): lane L holds (M=L%16, K=2*(L>=16)) in vgpr0 and K+1
// in vgpr1 -> lane L loads elements 2L and 2L+1 of each 64-value tile.
// C/D layout: lane0 vgpr0..7 = M0..7 @ N0, lane16 vgpr0..7 = M8..15 @ N0.

__global__ void finalize_kernel(const float* __restrict__ ws,
                                const float* __restrict__ slog,
                                const int*   __restrict__ slab,
                                float* __restrict__ out) {
  const int lane = threadIdx.x;   // 32 threads, one wave

  // sentiment masked CE over 256 rows of 3 logits
  float ces = 0.f, ns = 0.f;
  for (int b = lane; b < kB; b += 32) {
    int lab = slab[b];
    bool valid = (lab != IGNORE_IDX);
    int safe = valid ? lab : 0;
    float l0 = slog[3*b], l1 = slog[3*b + 1], l2 = slog[3*b + 2];
    float m = fmaxf(l0, fmaxf(l1, l2));
    float lse = m + __logf(__expf(l0-m) + __expf(l1-m) + __expf(l2-m));
    float lt = (safe == 0) ? l0 : ((safe == 1) ? l1 : l2);
    if (valid) { ces += lse - lt; ns += 1.0f; }
  }
#pragma unroll
  for (int off = 16; off > 0; off >>= 1) {
    ces += __shfl_down(ces, off, 32);
    ns  += __shfl_down(ns,  off, 32);
  }
  ces = __shfl(ces, 0, 32);
  ns  = __shfl(ns,  0, 32);

  // WMMA-based block-partial reduction (EXEC all-ones: single full wave,
  // no divergence around the wmma ops)
  float tot[5];
  v2f ones = {1.0f, 1.0f};
#pragma unroll
  for (int a = 0; a < 5; ++a) {
    v8f c = {0.f, 0.f, 0.f, 0.f, 0.f, 0.f, 0.f, 0.f};
    const float* base = ws + a * kNB;
    for (int t = 0; t < kNB / 64; ++t) {
      v2f A;
      A.x = base[t * 64 + 2 * lane];
      A.y = base[t * 64 + 2 * lane + 1];
      c = __builtin_amdgcn_wmma_f32_16x16x4_f32(
          /*neg_a=*/false, A, /*neg_b=*/false, ones,
          /*c_mod=*/(short)0, c, /*reuse_a=*/false, /*reuse_b=*/false);
    }
    float s = c[0] + c[1] + c[2] + c[3] + c[4] + c[5] + c[6] + c[7];
    tot[a] = __shfl(s, 0, 32) + __shfl(s, 16, 32);  // column 0: M0..7 + M8..15
  }

  if (lane == 0) {
    float aspect   = (tot[1] > 0.f) ? tot[0] / fmaxf(tot[1], 1.0f) : 0.0f;
    float opinion  = (tot[3] > 0.f) ? tot[2] / fmaxf(tot[3], 1.0f) : 0.0f;
    float senti    = ces / fmaxf(ns, 1.0f);
    float boundary = tot[4] / (float)(2 * kN);
    out[0] = aspect + opinion + senti + 0.5f * boundary;
  }
}

// ---------------- launch ----------------

extern "C" void kernel_launch(void* const* d_in, const int* in_sizes, int n_in,
                              void* d_out, int out_size, void* d_ws, size_t ws_size,
                              hipStream_t stream) {
  const float* alog = (const float*)d_in[0];   // aspect_logits  [B,S,3]
  const float* olog = (const float*)d_in[1];   // opinion_logits [B,S,3]
  const float* slog = (const float*)d_in[2];   // sentiment_logits [B,3]
  const float* blog = (const float*)d_in[3];   // boundary_logits [B,S,2]
  const int*   alab = (const int*)d_in[4];     // aspect_labels [B,S]
  const int*   olab = (const int*)d_in[5];     // opinion_labels [B,S]
  const int*   slab = (const int*)d_in[6];     // sentiment_labels [B]
  float* out = (float*)d_out;
  float* ws  = (float*)d_ws;                   // 5 * kNB floats (20 KB)

  partial_kernel<<<kNB, kTPB, 0, stream>>>(alog, olog, blog, alab, olab, ws);
  finalize_kernel<<<1, 32, 0, stream>>>(ws, slog, slab, out);
}